// PeriodicPrimitives2D_7980049236370
// MI455X (gfx1250) — compile-verified
//
#include <hip/hip_runtime.h>

typedef float v2f __attribute__((ext_vector_type(2)));
typedef float v8f __attribute__((ext_vector_type(8)));

#define KMAX   32
#define GCHUNK 64

// ------------------------------------------------------------------
// Kernel A: per-(gaussian,dim) row, select top-K |coeff| of F values.
// One wave32 per row; each lane owns 32 coefficients in registers.
// Writes (signed_coeff, prescaled_frequency) pairs to sel[row][k].
// ------------------------------------------------------------------
__global__ void topk_select_kernel(const float* __restrict__ wc,
                                   const int*  __restrict__ ntop,
                                   const int*  __restrict__ nrand,
                                   const int*  __restrict__ maxf,
                                   float2* __restrict__ sel,
                                   int nrows, int F) {
  const int lane = threadIdx.x & 31;
  const int wave = (int)((blockIdx.x * blockDim.x + threadIdx.x) >> 5);
  if (wave >= nrows) return;

  const int   K      = min(ntop[0] + nrand[0], KMAX);
  const float fscale = (float)maxf[0] / (float)F;

  float v[32];
#pragma unroll
  for (int k = 0; k < 32; ++k) {
    int idx = k * 32 + lane;
    v[k] = (idx < F) ? wc[(long)wave * F + idx] : 0.0f;
  }

  unsigned chosen = 0u;
  for (int it = 0; it < K; ++it) {
    float bestA = -1.0f;
    float bestS = 0.0f;
    int   bestK = 0;
#pragma unroll
    for (int k = 0; k < 32; ++k) {
      if (!((chosen >> k) & 1u)) {
        float a = fabsf(v[k]);
        if (a > bestA) { bestA = a; bestS = v[k]; bestK = k; }
      }
    }
    // wave32 argmax reduction on (bestA, lane); ties -> lowest lane
    unsigned bl = (unsigned)lane;
#pragma unroll
    for (int off = 16; off >= 1; off >>= 1) {
      float    oa = __shfl_xor(bestA, off, 32);
      unsigned ol = __shfl_xor(bl,    off, 32);
      if (oa > bestA || (oa == bestA && ol < bl)) { bestA = oa; bl = ol; }
    }
    if ((unsigned)lane == bl) {
      int idx = bestK * 32 + lane;
      sel[(long)wave * KMAX + it] = make_float2(bestS, fscale * (float)idx);
      chosen |= (1u << bestK);
    }
  }
}

// ------------------------------------------------------------------
// Evaluate w for TWO gaussians (k0, k1) of one point at once.
// Four independent accumulator chains (wx0/wy0/wx1/wy1) keep the
// quarter-rate TRANS pipe (v_cos_f32) saturated without v_nop stalls.
// v_cos_f32 input is in units of 2*pi, so cos(2pi f t) = v_cos(f*t).
// ------------------------------------------------------------------
__device__ __forceinline__ void eval_pair(float2 xp,
                                          const float2* __restrict__ sPos,
                                          const float2* __restrict__ sScl,
                                          const float2* __restrict__ sCS,
                                          const float2* __restrict__ sSel,
                                          int k0, int k1, int K,
                                          float& w0, float& w1) {
  const float2 p0 = sPos[k0], p1 = sPos[k1];
  const float2 s0 = sScl[k0], s1 = sScl[k1];
  const float2 r0 = sCS[k0],  r1 = sCS[k1];
  const float dx0 = xp.x - p0.x, dy0 = xp.y - p0.y;
  const float dx1 = xp.x - p1.x, dy1 = xp.y - p1.y;
  const float tx0 = fmaf(dx0, r0.x,  dy0 * r0.y);
  const float ty0 = fmaf(-dx0, r0.y, dy0 * r0.x);
  const float tx1 = fmaf(dx1, r1.x,  dy1 * r1.y);
  const float ty1 = fmaf(-dx1, r1.y, dy1 * r1.x);
  const float ax0 = tx0 * s0.x, ay0 = ty0 * s0.y;
  const float ax1 = tx1 * s1.x, ay1 = ty1 * s1.y;
  const float e0 = __expf(-0.5f * fmaf(ax0, ax0, ay0 * ay0));
  const float e1 = __expf(-0.5f * fmaf(ax1, ax1, ay1 * ay1));

  const float2* __restrict__ sx0 = &sSel[(k0 * 2 + 0) * KMAX];
  const float2* __restrict__ sy0 = &sSel[(k0 * 2 + 1) * KMAX];
  const float2* __restrict__ sx1 = &sSel[(k1 * 2 + 0) * KMAX];
  const float2* __restrict__ sy1 = &sSel[(k1 * 2 + 1) * KMAX];

  float wx0 = 0.f, wy0 = 0.f, wx1 = 0.f, wy1 = 0.f;
  if (K == 16) {               // uniform fast path (reference default: 8+8)
#pragma unroll
    for (int k = 0; k < 16; ++k) {
      float2 a0 = sx0[k], b0 = sy0[k], a1 = sx1[k], b1 = sy1[k];
      wx0 = fmaf(a0.x, __builtin_amdgcn_cosf(a0.y * tx0), wx0);
      wy0 = fmaf(b0.x, __builtin_amdgcn_cosf(b0.y * ty0), wy0);
      wx1 = fmaf(a1.x, __builtin_amdgcn_cosf(a1.y * tx1), wx1);
      wy1 = fmaf(b1.x, __builtin_amdgcn_cosf(b1.y * ty1), wy1);
    }
  } else {                     // generic fallback, still 4 streams
    for (int k = 0; k < K; ++k) {
      float2 a0 = sx0[k], b0 = sy0[k], a1 = sx1[k], b1 = sy1[k];
      wx0 = fmaf(a0.x, __builtin_amdgcn_cosf(a0.y * tx0), wx0);
      wy0 = fmaf(b0.x, __builtin_amdgcn_cosf(b0.y * ty0), wy0);
      wx1 = fmaf(a1.x, __builtin_amdgcn_cosf(a1.y * tx1), wx1);
      wy1 = fmaf(b1.x, __builtin_amdgcn_cosf(b1.y * ty1), wy1);
    }
  }
  w0 = e0 * wx0 * wy0;
  w1 = e1 * wx1 * wy1;
}

// ------------------------------------------------------------------
// Kernel B: render. 8 waves/block, 16 points/wave.
// Gaussian loop in super-chunks of 64 staged through LDS; inner
// K-chunks of 4 gaussians accumulated with V_WMMA_F32_16X16X4_F32:
//   D[16 pts][16 cols] += A[16 pts][4 g] x B[4 g][16 cols(RGB,0...)]
// ------------------------------------------------------------------
__global__ __launch_bounds__(256)
void render_kernel(const float* __restrict__ x,
                   const float* __restrict__ colors,
                   const float* __restrict__ pos,
                   const float* __restrict__ scales,
                   const float* __restrict__ rots,
                   const float2* __restrict__ sel,
                   const int* __restrict__ ntop,
                   const int* __restrict__ nrand,
                   float* __restrict__ out,
                   int N, int G) {
  __shared__ float2 sPos[GCHUNK];
  __shared__ float2 sScl[GCHUNK];
  __shared__ float2 sCS[GCHUNK];
  __shared__ float  sColPad[GCHUNK * 4];       // column-padded RGB0
  __shared__ float2 sSel[GCHUNK * 2 * KMAX];   // 32 KB

  const int t    = threadIdx.x;
  const int lane = t & 31;
  const int wave = t >> 5;
  const int hi   = lane >> 4;       // which half of the wave
  const int n    = lane & 15;       // B/C column (0..2 = RGB)
  const int cidx = (n < 3) ? n : 3; // padded color slot (slot 3 == 0)
  const int K    = min(ntop[0] + nrand[0], KMAX);

  const int pbase = blockIdx.x * 128 + wave * 16;
  const int pIdx  = min(pbase + n, N - 1);
  const float2 xp = make_float2(x[2 * pIdx], x[2 * pIdx + 1]);

  v8f acc = {};

  const int nSC = (G + GCHUNK - 1) / GCHUNK;
  for (int sc = 0; sc < nSC; ++sc) {
    const int gbase = sc * GCHUNK;
    __syncthreads();                 // protect LDS reuse across iterations
    // ---- stage per-gaussian params ----
    for (int g = t; g < GCHUNK; g += 256) {
      int gg = gbase + g;
      if (gg < G) {
        sPos[g] = make_float2(pos[2 * gg], pos[2 * gg + 1]);
        sScl[g] = make_float2(scales[2 * gg], scales[2 * gg + 1]);
        float sr, cr;
        __sincosf(rots[gg], &sr, &cr);
        sCS[g] = make_float2(cr, sr);
        sColPad[g * 4 + 0] = colors[3 * gg];
        sColPad[g * 4 + 1] = colors[3 * gg + 1];
        sColPad[g * 4 + 2] = colors[3 * gg + 2];
        sColPad[g * 4 + 3] = 0.0f;
      } else {
        sPos[g] = make_float2(0.f, 0.f);
        sScl[g] = make_float2(0.f, 0.f);
        sCS[g]  = make_float2(1.f, 0.f);
        sColPad[g * 4 + 0] = 0.0f;
        sColPad[g * 4 + 1] = 0.0f;
        sColPad[g * 4 + 2] = 0.0f;
        sColPad[g * 4 + 3] = 0.0f;
      }
    }
    // ---- stage selected-frequency table (float4 vectorized) ----
    {
      const float4* g4 = (const float4*)sel;
      float4*       s4 = (float4*)sSel;
      const int f4PerRow = KMAX / 2;                 // 16 float4 per row
      const int total    = GCHUNK * 2 * f4PerRow;    // 2048
      const int gb4      = gbase * 2 * f4PerRow;
      for (int i = t; i < total; i += 256) {
        float4 val = make_float4(0.f, 0.f, 0.f, 0.f);
        if ((gbase * 2 + (i / f4PerRow)) < 2 * G) val = g4[gb4 + i];
        s4[i] = val;
      }
    }
    __syncthreads();
    // ---- accumulate 16 WMMA K-chunks of 4 gaussians each ----
    for (int c = 0; c < GCHUNK / 4; ++c) {
      // A-matrix 16x4 f32 layout: VGPR0 holds K=0(lo)/K=2(hi),
      //                           VGPR1 holds K=1(lo)/K=3(hi)
      const int k0 = 4 * c + (hi ? 2 : 0);
      const int k1 = 4 * c + (hi ? 3 : 1);
      float w0, w1;
      eval_pair(xp, sPos, sScl, sCS, sSel, k0, k1, K, w0, w1);
      const float b0 = sColPad[k0 * 4 + cidx];
      const float b1 = sColPad[k1 * 4 + cidx];
      v2f a; a[0] = w0; a[1] = w1;
      v2f b; b[0] = b0; b[1] = b1;
      acc = __builtin_amdgcn_wmma_f32_16x16x4_f32(
                false, a, false, b, (short)0, acc, false, false);
    }
  }

  // ---- store C tile: VGPR v -> row (v + 8*hi), col n; cols 0..2 valid ----
  if (n < 3) {
    const int rowbase = pbase + (hi ? 8 : 0);
#pragma unroll
    for (int v = 0; v < 8; ++v) {
      int pt = rowbase + v;
      if (pt < N) out[3 * pt + n] = acc[v];
    }
  }
}

// ------------------------------------------------------------------
extern "C" void kernel_launch(void* const* d_in, const int* in_sizes, int n_in,
                              void* d_out, int out_size, void* d_ws, size_t ws_size,
                              hipStream_t stream) {
  const float* x      = (const float*)d_in[0];
  const float* colors = (const float*)d_in[1];
  const float* pos    = (const float*)d_in[2];
  const float* scales = (const float*)d_in[3];
  const float* rots   = (const float*)d_in[4];
  const float* wc     = (const float*)d_in[5];
  const int*   ntop   = (const int*)d_in[6];
  const int*   nrand  = (const int*)d_in[7];
  const int*   maxf   = (const int*)d_in[8];

  const int N = in_sizes[0] / 2;          // 16384 points
  const int G = in_sizes[2] / 2;          // 512 gaussians
  const int F = in_sizes[5] / (2 * G);    // 1024 frequencies
  const int nrows = 2 * G;

  float2* sel = (float2*)d_ws;            // nrows * KMAX float2 = 256 KB

  {
    const int wavesPerBlock = 8;          // 256 threads
    const int blocks = (nrows + wavesPerBlock - 1) / wavesPerBlock;
    topk_select_kernel<<<blocks, 256, 0, stream>>>(wc, ntop, nrand, maxf,
                                                   sel, nrows, F);
  }
  {
    const int blocks = (N + 127) / 128;   // 128 points per block
    render_kernel<<<blocks, 256, 0, stream>>>(x, colors, pos, scales, rots,
                                              sel, ntop, nrand,
                                              (float*)d_out, N, G);
  }
}